// GNNLayer_66254165508714
// MI455X (gfx1250) — compile-verified
//
#include <hip/hip_runtime.h>
#include <cstdint>
#include <cstddef>

typedef __attribute__((ext_vector_type(2))) float v2f;
typedef __attribute__((ext_vector_type(4))) float v4f;
typedef __attribute__((ext_vector_type(8))) float v8f;

static constexpr int B = 2, N = 512, H = 256, E = 64, M = 256, O = 256;

// ---------------------------------------------------------------------------
// Kernel 1: esum[b,i,:] = sum_j mask[b,j] * e[b,i,j,:]
// The only bandwidth-heavy pass: streams all of e (134 MB) once, coalesced
// b128 non-temporal loads (no reuse -> keep L2 clean for weights).
// ---------------------------------------------------------------------------
__global__ void __launch_bounds__(256) k_esum(const float* __restrict__ e,
                                              const uint8_t* __restrict__ mask,
                                              float* __restrict__ esum) {
  const int bi  = blockIdx.x;          // b*N + i  (1024 blocks)
  const int b   = bi >> 9;
  const int tid = threadIdx.x;
  const int k4  = tid & 15;            // which float4 of the 64-float row
  const int jg  = tid >> 4;            // j-group 0..15
  const float* ebase = e + (size_t)bi * (size_t)(N * E);

  v4f acc = {0.f, 0.f, 0.f, 0.f};
#pragma unroll 4
  for (int j = jg; j < N; j += 16) {
    const float mf = mask[(b << 9) + j] ? 1.0f : 0.0f;
    const v4f ev = __builtin_nontemporal_load(
        (const v4f*)(ebase + (size_t)j * E + k4 * 4));
    acc += ev * mf;
  }

  __shared__ v4f red[256];
  red[tid] = acc;
  __syncthreads();
#pragma unroll
  for (int s = 8; s > 0; s >>= 1) {
    if (jg < s) red[tid] += red[tid + s * 16];
    __syncthreads();
  }
  if (jg == 0) *(v4f*)(esum + (size_t)bi * E + k4 * 4) = red[tid];
}

// ---------------------------------------------------------------------------
// Kernel 2: per-batch cnt_b = sum_j mask, mjsum[b,:] = (sum_j mask_j h[b,j,:]) @ Wmj
// ---------------------------------------------------------------------------
__global__ void __launch_bounds__(256) k_prep(const float* __restrict__ h,
                                              const uint8_t* __restrict__ mask,
                                              const float* __restrict__ Wm,
                                              float* __restrict__ cnt_ws,
                                              float* __restrict__ mjsum_ws) {
  const int b = blockIdx.x;
  const int t = threadIdx.x;           // 256 threads
  __shared__ float hs[H];

  float acc = 0.f;
  for (int j = 0; j < N; ++j) {
    const float mf = mask[b * N + j] ? 1.0f : 0.0f;
    acc += mf * h[((size_t)(b * N + j)) * H + t];
  }
  hs[t] = acc;
  if (t == 0) {
    float c = 0.f;
    for (int j = 0; j < N; ++j) c += mask[b * N + j] ? 1.0f : 0.0f;
    cnt_ws[b] = c;
  }
  __syncthreads();

  const float* Wmj = Wm + (size_t)H * M;   // rows H..2H-1 of Wm
  float a2 = 0.f;
  for (int hh = 0; hh < H; ++hh) a2 += hs[hh] * Wmj[(size_t)hh * M + t];
  mjsum_ws[b * M + t] = a2;
}

// ---------------------------------------------------------------------------
// Kernel 3: fused WMMA GEMMs.  One workgroup per 16-row strip (64 WGs).
//   Phase 1: msum = mask_i*(cnt*(h@Wmi + bm) + mjsum + esum@Wme)  -> LDS
//   Phase 2: o    = h@Wf[:H] + msum@Wf[H:] + bf                   -> global
// Uses V_WMMA_F32_16X16X4_F32 (fp32 in/out, wave32).
// A 16x4 f32 layout: lane%16 = row M; lanes>=16 take K+2,K+3 (VGPR0/1 = K,K+1).
// C/D  layout: M = vgpr + 8*(lane>=16), N = lane%16.
// ---------------------------------------------------------------------------
#define HS_LD 260   // padded LDS strides (word stride % 64 == 4 -> conflict-free)
#define ES_LD 68
#define MS_LD 260

__global__ void __launch_bounds__(256) k_fused(const float* __restrict__ h,
                                               const uint8_t* __restrict__ mask,
                                               const float* __restrict__ Wm,
                                               const float* __restrict__ bm,
                                               const float* __restrict__ Wf,
                                               const float* __restrict__ bf,
                                               const float* __restrict__ esum,
                                               const float* __restrict__ cnt_ws,
                                               const float* __restrict__ mjsum_ws,
                                               float* __restrict__ out) {
  const int tid  = threadIdx.x;
  const int wave = tid >> 5;
  const int lane = tid & 31;
  const int row0 = blockIdx.x * 16;    // flattened row b*N+i
  const int b    = row0 >> 9;

  __shared__ float h_s[16 * HS_LD];
  __shared__ float es_s[16 * ES_LD];
  __shared__ float ms_s[16 * MS_LD];
  __shared__ float mask_s[16];
  __shared__ float cnt_sh;

  // ---- stage h strip (16x256) and esum strip (16x64), coalesced b128 ----
  for (int idx = tid * 4; idx < 16 * H; idx += 256 * 4) {
    const int r = idx >> 8, c = idx & 255;
    const v4f v = *(const v4f*)(h + (size_t)(row0 + r) * H + c);
    *(v4f*)(h_s + r * HS_LD + c) = v;
  }
  {
    const int idx = tid * 4;                       // 256*4 == 16*64 exactly
    const int r = idx >> 6, c = idx & 63;
    const v4f v = *(const v4f*)(esum + (size_t)(row0 + r) * E + c);
    *(v4f*)(es_s + r * ES_LD + c) = v;
  }
  if (tid < 16) mask_s[tid] = mask[row0 + tid] ? 1.0f : 0.0f;
  if (tid == 0) cnt_sh = cnt_ws[b];
  __syncthreads();

  const float cnt   = cnt_sh;
  const int   r16   = lane & 15;
  const int   khalf = (lane >> 4) << 1;            // 0 or 2
  const int   rhalf = (lane >> 4) << 3;            // 0 or 8 (C/D row offset)

  const float* Wmi = Wm;                           // rows 0..H-1
  const float* Wme = Wm + (size_t)(2 * H) * M;     // rows 2H..2H+E-1

  // -------------------- Phase 1: msum strip -> LDS --------------------
  for (int t = wave; t < 16; t += 8) {             // uniform per wave (2 tiles)
    const int col0 = t << 4;
    const int n    = col0 + r16;
    v8f acc = {0.f, 0.f, 0.f, 0.f, 0.f, 0.f, 0.f, 0.f};

    for (int k = 0; k < H; k += 4) {               // h @ Wmi
      const int kk = k + khalf;
      const v2f a = *(const v2f*)(h_s + r16 * HS_LD + kk);
      v2f bb;
      bb.x = Wmi[(size_t)kk * M + n];
      bb.y = Wmi[(size_t)(kk + 1) * M + n];
      acc = __builtin_amdgcn_wmma_f32_16x16x4_f32(false, a, false, bb,
                                                  (short)0, acc, false, false);
    }
#pragma unroll
    for (int v = 0; v < 8; ++v) acc[v] *= cnt;     // cnt * mi

    for (int k = 0; k < E; k += 4) {               // + esum @ Wme
      const int kk = k + khalf;
      const v2f a = *(const v2f*)(es_s + r16 * ES_LD + kk);
      v2f bb;
      bb.x = Wme[(size_t)kk * M + n];
      bb.y = Wme[(size_t)(kk + 1) * M + n];
      acc = __builtin_amdgcn_wmma_f32_16x16x4_f32(false, a, false, bb,
                                                  (short)0, acc, false, false);
    }

    const float colv = cnt * bm[n] + mjsum_ws[b * M + n];
#pragma unroll
    for (int v = 0; v < 8; ++v) {
      const int rowM = v + rhalf;
      ms_s[rowM * MS_LD + n] = mask_s[rowM] * (acc[v] + colv);
    }
  }
  __syncthreads();

  // -------------------- Phase 2: o strip -> global --------------------
  const float* Wfh = Wf;                           // rows 0..H-1
  const float* Wfm = Wf + (size_t)H * O;           // rows H..H+M-1

  for (int t = wave; t < 16; t += 8) {
    const int col0 = t << 4;
    const int n    = col0 + r16;
    v8f acc = {0.f, 0.f, 0.f, 0.f, 0.f, 0.f, 0.f, 0.f};

    for (int k = 0; k < H; k += 4) {               // h @ Wf[:H]
      const int kk = k + khalf;
      const v2f a = *(const v2f*)(h_s + r16 * HS_LD + kk);
      v2f bb;
      bb.x = Wfh[(size_t)kk * O + n];
      bb.y = Wfh[(size_t)(kk + 1) * O + n];
      acc = __builtin_amdgcn_wmma_f32_16x16x4_f32(false, a, false, bb,
                                                  (short)0, acc, false, false);
    }
    for (int k = 0; k < M; k += 4) {               // + msum @ Wf[H:]
      const int kk = k + khalf;
      const v2f a = *(const v2f*)(ms_s + r16 * MS_LD + kk);
      v2f bb;
      bb.x = Wfm[(size_t)kk * O + n];
      bb.y = Wfm[(size_t)(kk + 1) * O + n];
      acc = __builtin_amdgcn_wmma_f32_16x16x4_f32(false, a, false, bb,
                                                  (short)0, acc, false, false);
    }

    const float bfv = bf[n];
#pragma unroll
    for (int v = 0; v < 8; ++v) {
      const int rowM = v + rhalf;
      out[(size_t)(row0 + rowM) * O + n] = acc[v] + bfv;
    }
  }
}

// ---------------------------------------------------------------------------
extern "C" void kernel_launch(void* const* d_in, const int* in_sizes, int n_in,
                              void* d_out, int out_size, void* d_ws, size_t ws_size,
                              hipStream_t stream) {
  const float*   h    = (const float*)d_in[0];
  const float*   e    = (const float*)d_in[1];
  const uint8_t* mask = (const uint8_t*)d_in[2];   // jax bool -> 1 byte/elem
  const float*   Wm   = (const float*)d_in[3];
  const float*   bm   = (const float*)d_in[4];
  const float*   Wf   = (const float*)d_in[5];
  const float*   bf   = (const float*)d_in[6];
  float*         out  = (float*)d_out;

  // workspace layout (floats): esum[B*N*E] | mjsum[B*M] | cnt[B]
  float* esum  = (float*)d_ws;
  float* mjsum = esum + (size_t)B * N * E;
  float* cnt   = mjsum + (size_t)B * M;

  k_esum <<<B * N,      256, 0, stream>>>(e, mask, esum);
  k_prep <<<B,          256, 0, stream>>>(h, mask, Wm, cnt, mjsum);
  k_fused<<<(B * N)/16, 256, 0, stream>>>(h, mask, Wm, bm, Wf, bf,
                                          esum, cnt, mjsum, out);
}